// MessagePassing_28252294873249
// MI455X (gfx1250) — compile-verified
//
#include <hip/hip_runtime.h>

// out[i] = sum_{e: dst[e]==i} x[src[e]]   (segment-sum scatter-add)
// x: [N,64] f32, edge_index: [2,E] int (row0=dst, row1=src), out: [N,64] f32
//
// MI455X strategy: memory/atomic-bound. x (25.6MB) and out (25.6MB) are
// L2-resident (192MB L2). Hot loop per edge: one coalesced 256B gather
// (wave32, float2/lane) + 64 no-return global_atomic_add_f32 resolved at the
// L2 atomic units. Edge indices are double-buffer staged into LDS with the
// gfx1250 async-to-LDS engine (ASYNCcnt) so index fetch latency overlaps the
// gather/atomic work of the previous tile; per-edge indices are broadcast to
// SGPRs with v_readlane.
//
// (scatter_add_kernel is defined first in this TU so its device asm appears
// in the disassembly snippet.)

#define WAVES_PER_BLOCK 8
#define BLOCK_THREADS   (WAVES_PER_BLOCK * 32)
#define TILE            32   // edges per wave per pipeline stage
#define NTILES          2    // pipeline stages per wave => 64 edges/wave

typedef __attribute__((address_space(1))) int as1_int;   // global
typedef __attribute__((address_space(3))) int as3_int;   // LDS

__device__ __forceinline__ void async_idx_b32(const int* gsrc, int* lds_dst) {
#if __has_builtin(__builtin_amdgcn_global_load_async_to_lds_b32)
    // gfx1250 async DMA: copies 4 bytes from global address gsrc to LDS address lds_dst, tracked by ASYNCcnt.
    __builtin_amdgcn_global_load_async_to_lds_b32(
        (as1_int*)gsrc, (as3_int*)lds_dst, /*imm offset*/0, /*cpol*/0);
#else
    *lds_dst = *gsrc;  // synchronous fallback (ASYNCcnt stays 0; waits are no-ops)
#endif
}

// Full compiler memory barrier + hardware wait: LDS reads of staged indices
// must not be hoisted above this point.
#define WAIT_ASYNCCNT(N) asm volatile("s_wait_asynccnt %0" :: "n"(N) : "memory")

__device__ __forceinline__ void stage_tile(const int* __restrict__ ei, long long ebase,
                                           int E, int lane, int* ldsD, int* ldsS) {
    long long e = ebase + lane;
    if (e >= E) e = (long long)E - 1;  // clamp: duplicate fetch, never consumed
    async_idx_b32(ei + e,                 ldsD + lane);
    async_idx_b32(ei + (long long)E + e,  ldsS + lane);
}

__device__ __forceinline__ void process_tile(const float* __restrict__ x,
                                             float* __restrict__ out,
                                             const int* ldsD, const int* ldsS,
                                             int lane, long long ebase, int E) {
    // Each lane holds one edge's (dst,src); broadcast per edge via v_readlane
    // so row base addresses are scalar and the gather stays one load/edge.
    const int dv = ldsD[lane];
    const int sv = ldsS[lane];
    const long long rem = (long long)E - ebase;
    const int nvalid = rem < TILE ? (int)rem : TILE;
#pragma unroll 8
    for (int k = 0; k < TILE; ++k) {
        if (k >= nvalid) break;
        const int dst = __builtin_amdgcn_readlane(dv, k);
        const int src = __builtin_amdgcn_readlane(sv, k);

        // Coalesced 256B row gather: lane l takes floats [2l, 2l+1].
        const float2 v = ((const float2* __restrict__)(x + (long long)src * 64))[lane];

        float* o = out + (long long)dst * 64 + (lane << 1);
        unsafeAtomicAdd(o,     v.x);   // -> global_atomic_add_f32 (no return)
        unsafeAtomicAdd(o + 1, v.y);
    }
}

__global__ __launch_bounds__(BLOCK_THREADS) void scatter_add_kernel(
    const float* __restrict__ x,
    const int*   __restrict__ ei,   // [2*E]: ei[e]=dst, ei[E+e]=src
    float*       __restrict__ out,
    int E) {
    __shared__ int lds_idx[WAVES_PER_BLOCK][2][2 * TILE];  // [wave][buf][dst|src]

    const int lane = threadIdx.x & 31;
    const int wid  = __builtin_amdgcn_readfirstlane((int)(threadIdx.x >> 5));
    const long long gwave = (long long)blockIdx.x * WAVES_PER_BLOCK + wid;
    const long long base  = gwave * (TILE * NTILES);
    if (base >= E) return;

    // Stream the linearly-consumed index arrays ahead (global_prefetch_b8).
    __builtin_prefetch(ei + base + 16384, 0, 0);
    __builtin_prefetch(ei + (long long)E + base + 16384, 0, 0);

    int* buf0 = lds_idx[wid][0];
    int* buf1 = lds_idx[wid][1];

    // Prologue: stage tile 0 (2 async instructions outstanding).
    stage_tile(ei, base, E, lane, buf0, buf0 + TILE);

#pragma unroll
    for (int t = 0; t < NTILES; ++t) {
        const long long ebase = base + (long long)t * TILE;
        if (ebase >= E) break;
        int* cur = (t & 1) ? buf1 : buf0;
        int* nxt = (t & 1) ? buf0 : buf1;

        const long long nebase = ebase + TILE;
        if (t + 1 < NTILES && nebase < E) {
            // Issue tile t+1 (counter -> 4), then wait for tile t (<=2).
            stage_tile(ei, nebase, E, lane, nxt, nxt + TILE);
            WAIT_ASYNCCNT(2);
        } else {
            WAIT_ASYNCCNT(0);
        }
        process_tile(x, out, cur, cur + TILE, lane, ebase, E);
    }
}

__global__ __launch_bounds__(256) void zero_fill_f4(float4* __restrict__ out, int n4) {
    int i = blockIdx.x * blockDim.x + threadIdx.x;
    if (i < n4) {
        out[i] = make_float4(0.f, 0.f, 0.f, 0.f);
    }
}

extern "C" void kernel_launch(void* const* d_in, const int* in_sizes, int n_in,
                              void* d_out, int out_size, void* d_ws, size_t ws_size,
                              hipStream_t stream) {
    const float* x   = (const float*)d_in[0];
    const int*   ei  = (const int*)d_in[1];
    float*       out = (float*)d_out;

    const int E  = in_sizes[1] / 2;      // 1,600,000
    const int n4 = out_size / 4;         // N*64 floats, divisible by 4

    zero_fill_f4<<<(n4 + 255) / 256, 256, 0, stream>>>((float4*)out, n4);

    const long long edges_per_wave = (long long)TILE * NTILES;          // 64
    const long long waves  = ((long long)E + edges_per_wave - 1) / edges_per_wave;
    const int       blocks = (int)((waves + WAVES_PER_BLOCK - 1) / WAVES_PER_BLOCK);
    scatter_add_kernel<<<blocks, BLOCK_THREADS, 0, stream>>>(x, ei, out, E);
}